// HarmonicFeatureExtractor_5738076307836
// MI455X (gfx1250) — compile-verified
//
#include <hip/hip_runtime.h>
#include <math.h>

typedef float v8f __attribute__((ext_vector_type(8)));
typedef float v2f __attribute__((ext_vector_type(2)));

#define N_FFT      2048
#define HOP        512
#define NBINS      1025
#define NBINS_PAD  1040      /* 65 * 16 */
#define NTILE_M    65
#define AUDIO_LEN  655360
#define NFRAMES    1281
#define FRAME_TILES 81
#define BATCH      32
#define CHUNK      256
#define PI_F       3.14159265358979323846f

// ---------------------------------------------------------------------------
// Build DFT tables: cosT[bin][k], sinT[bin][k]; bins >= 1025 zeroed (padding).
// Phase computed mod 2048 so cosf/sinf see small arguments.
// ---------------------------------------------------------------------------
__global__ void build_dft_kernel(float* __restrict__ cosT,
                                 float* __restrict__ sinT) {
    int idx = blockIdx.x * blockDim.x + threadIdx.x;
    if (idx >= NBINS_PAD * N_FFT) return;
    int bin = idx >> 11;
    int k   = idx & (N_FFT - 1);
    float c = 0.0f, s = 0.0f;
    if (bin < NBINS) {
        int phase = (bin * k) & (N_FFT - 1);
        float ang = (2.0f * PI_F / (float)N_FFT) * (float)phase;
        __sincosf(ang, &s, &c);
    }
    cosT[idx] = c;
    sinT[idx] = s;
}

// ---------------------------------------------------------------------------
// Fused: DFT-as-WMMA-GEMM -> |spec| in LDS -> argmax -> harmonic log-energy.
// One workgroup = 16 frames (one frame tile) of one batch.
// 512 threads = 16 wave32; wave w owns bin-tiles {w, w+16, w+32, w+48, w+64}.
// ---------------------------------------------------------------------------
__launch_bounds__(512)
__global__ void harmonic_kernel(const float* __restrict__ audio,
                                const float* __restrict__ cosT,
                                const float* __restrict__ sinT,
                                float* __restrict__ out) {
    // Windowed frame chunk, float2 K-pair layout: [k/2][n] pairs (k even, k odd)
    __shared__ float lfr[(CHUNK / 2) * 32];            // 16 KB
    // Full magnitude column for this frame tile: [bin][frame]
    __shared__ float lmag[NBINS_PAD * 16];             // 66.5 KB

    const int tileT  = blockIdx.x;          // frame tile 0..80
    const int b      = blockIdx.y;          // batch
    const int tid    = threadIdx.x;
    const int wave   = tid >> 5;            // 0..15
    const int lane   = tid & 31;
    const int hi     = lane >> 4;           // K-half selector
    const int lm     = lane & 15;           // M (A rows) / N (B,C cols)
    const int frame0 = tileT * 16;

    const float* aud = audio + (size_t)b * AUDIO_LEN;

    const v8f vzero = {0.f, 0.f, 0.f, 0.f, 0.f, 0.f, 0.f, 0.f};
    v8f acc_re[5], acc_im[5];
#pragma unroll
    for (int j = 0; j < 5; ++j) { acc_re[j] = vzero; acc_im[j] = vzero; }

    for (int kc = 0; kc < N_FFT; kc += CHUNK) {
        __syncthreads();
        // ---- cooperative fill: reflect-padded, Hann-windowed frame chunk ----
#pragma unroll
        for (int r = 0; r < (CHUNK * 16) / 512; ++r) {
            int idx = tid + r * 512;
            int n   = idx & 15;             // frame within tile
            int kl  = idx >> 4;             // k within chunk
            int k   = kc + kl;
            int t   = frame0 + n;
            float v = 0.0f;
            if (t < NFRAMES) {
                int s = t * HOP + k - (N_FFT / 2);
                s = (s < 0) ? -s : s;
                s = (s >= AUDIO_LEN) ? (2 * AUDIO_LEN - 2 - s) : s;
                float w = 0.5f * (1.0f - __cosf((2.0f * PI_F / (float)N_FFT) * (float)k));
                v = aud[s] * w;
            }
            lfr[(kl >> 1) * 32 + n * 2 + (kl & 1)] = v;
        }
        __syncthreads();

        // ---- WMMA accumulation over this K chunk ----
#pragma unroll
        for (int j = 0; j < 5; ++j) {
            const int tileM = wave + 16 * j;        // wave-uniform guard
            if (tileM < NTILE_M) {
                const int m = tileM * 16 + lm;      // A row (bin)
                const float* cRow = cosT + (size_t)m * N_FFT + kc;
                const float* sRow = sinT + (size_t)m * N_FFT + kc;
#pragma unroll 4
                for (int kk = 0; kk < CHUNK; kk += 4) {
                    const int k0 = kk + hi * 2;
                    v2f a_c = *(const v2f*)(cRow + k0);
                    v2f a_s = *(const v2f*)(sRow + k0);
                    v2f bb  = *(const v2f*)(&lfr[(k0 >> 1) * 32 + lm * 2]);
                    acc_re[j] = __builtin_amdgcn_wmma_f32_16x16x4_f32(
                        false, a_c, false, bb, (short)0, acc_re[j], false, false);
                    acc_im[j] = __builtin_amdgcn_wmma_f32_16x16x4_f32(
                        false, a_s, false, bb, (short)0, acc_im[j], false, false);
                }
            }
        }
    }

    // ---- magnitudes into LDS ----
    __syncthreads();
#pragma unroll
    for (int j = 0; j < 5; ++j) {
        const int tileM = wave + 16 * j;
        if (tileM < NTILE_M) {
#pragma unroll
            for (int r = 0; r < 8; ++r) {
                int binRow = tileM * 16 + r + 8 * hi;   // C/D layout: M = vgpr + 8*(lane/16)
                float re = acc_re[j][r];
                float im = acc_im[j][r];
                lmag[binRow * 16 + lm] = sqrtf(re * re + im * im);
            }
        }
    }
    __syncthreads();

    // ---- argmax over bins (wave w handles frame w), first-max tie-break ----
    const int f = wave;
    float bestV = -1.0f;
    int   bestI = 0;
    for (int bin = lane; bin < NBINS; bin += 32) {
        float v = lmag[bin * 16 + f];
        if (v > bestV) { bestV = v; bestI = bin; }
    }
#pragma unroll
    for (int off = 16; off > 0; off >>= 1) {
        float ov = __shfl_xor(bestV, off, 32);
        int   oi = __shfl_xor(bestI, off, 32);
        if (ov > bestV || (ov == bestV && oi < bestI)) { bestV = ov; bestI = oi; }
    }

    // ---- harmonic gather + log10, lanes 0..4 are harmonics 1..5 ----
    const int t = frame0 + f;
    if (lane < 5 && t < NFRAMES) {
        int hb = bestI * (lane + 1);
        hb = (hb > NBINS - 1) ? (NBINS - 1) : hb;
        float e = lmag[hb * 16 + f];
        out[((size_t)b * 5 + (size_t)lane) * NFRAMES + t] = log10f(e + 1e-10f);
    }
}

// ---------------------------------------------------------------------------
extern "C" void kernel_launch(void* const* d_in, const int* in_sizes, int n_in,
                              void* d_out, int out_size, void* d_ws, size_t ws_size,
                              hipStream_t stream) {
    const float* audio = (const float*)d_in[0];
    float* out  = (float*)d_out;
    float* cosT = (float*)d_ws;                                   // 8.52 MB
    float* sinT = cosT + (size_t)NBINS_PAD * N_FFT;               // 8.52 MB

    int total = NBINS_PAD * N_FFT;
    build_dft_kernel<<<(total + 255) / 256, 256, 0, stream>>>(cosT, sinT);

    dim3 grid(FRAME_TILES, BATCH);
    harmonic_kernel<<<grid, 512, 0, stream>>>(audio, cosT, sinT, out);
}